// SpectralConv1d_86612310491733
// MI455X (gfx1250) — compile-verified
//
#include <hip/hip_runtime.h>
#include <math.h>

// Problem sizes fixed by the reference
#define NB     64      // batch
#define CIN    128     // in channels
#define COUT   128     // out channels
#define NMODE  64      // kept modes
#define NN     4096    // spatial length
#define LOG2N  12
#define KSPLIT 8       // K-split of the forward DHT GEMM

typedef float v2f __attribute__((ext_vector_type(2)));
typedef float v8f __attribute__((ext_vector_type(8)));

static __device__ __forceinline__ v8f wmma_f32_16x16x4(v2f a, v2f b, v8f c) {
  // D = A(16x4) x B(4x16) + C(16x16), fp32 accumulate
  return __builtin_amdgcn_wmma_f32_16x16x4_f32(false, a, false, b, (short)0, c,
                                               false, false);
}

// cas(2*pi*p/NN) with exact integer phase reduction (p mod NN)
static __device__ __forceinline__ float cas_tw(int p) {
  float th = (float)(p & (NN - 1)) * (6.28318530717958647692f / (float)NN);
  float s, c;
  sincosf(th, &s, &c);
  return c + s;
}

// ---------------------------------------------------------------------------
// Twiddle tables for the reference's recursive transform:
//   T[j,k] = prod over set bits b of j of cas(2*pi*k*2^b / NN)
// c1t[m][j] = T[j, m]          (m < 64, j < 4096)   -- forward B operand
// c2t[n][k] = T[k, n] / NN     (n < 4096, k < 64)   -- inverse B operand
// ---------------------------------------------------------------------------
__global__ void prep_cas(float* __restrict__ c1t, float* __restrict__ c2t) {
  int idx = blockIdx.x * blockDim.x + threadIdx.x;
  if (idx < NMODE * NN) {
    int m = idx >> LOG2N;
    int j = idx & (NN - 1);
    float v = 1.0f;
#pragma unroll
    for (int b = 0; b < LOG2N; ++b)
      if ((j >> b) & 1) v *= cas_tw(m << b);
    c1t[(size_t)m * NN + j] = v;
  } else {
    int t = idx - NMODE * NN;
    int n = t >> 6;
    int k = t & (NMODE - 1);
    float v = 1.0f / (float)NN;
#pragma unroll
    for (int b = 0; b < 6; ++b)   // k < 64 -> only low 6 bits
      if ((k >> b) & 1) v *= cas_tw(n << b);
    c2t[(size_t)n * NMODE + k] = v;
  }
}

// wp = w[k] + w[-k], wm = w[k] - w[-k]   (layout (i, o, k), k innermost)
__global__ void prep_w(const float* __restrict__ w, float* __restrict__ wp,
                       float* __restrict__ wm) {
  int idx = blockIdx.x * blockDim.x + threadIdx.x;
  if (idx >= CIN * COUT * NMODE) return;
  int k = idx & (NMODE - 1);
  int kneg = (NMODE - k) & (NMODE - 1);
  float a = w[idx];
  float b = w[idx - k + kneg];
  wp[idx] = a + b;
  wm[idx] = a - b;
}

// ---------------------------------------------------------------------------
// Forward: XHp[kc][r][m] = sum_{j in chunk kc} X[r][j] * c1t[m][j]
// One wave = 16 rows x 64 modes x (4096/KSPLIT) K. A/B fragments are b64 loads.
// ---------------------------------------------------------------------------
__global__ void dht_fwd(const float* __restrict__ x, const float* __restrict__ c1t,
                        float* __restrict__ xhp) {
  int wave = (blockIdx.x * blockDim.x + threadIdx.x) >> 5;
  int lane = threadIdx.x & 31;
  int rt = wave >> 3;              // 512 row tiles
  int kc = wave & (KSPLIT - 1);    // 8 K chunks
  int r0 = rt * 16;
  int mrow = lane & 15;
  int khalf = lane >> 4;           // K pair select per A/B layout
  const float* xr  = x   + (size_t)(r0 + mrow) * NN;
  const float* b0p = c1t + (size_t)(mrow)      * NN;
  const float* b1p = c1t + (size_t)(16 + mrow) * NN;
  const float* b2p = c1t + (size_t)(32 + mrow) * NN;
  const float* b3p = c1t + (size_t)(48 + mrow) * NN;
  int k0 = kc * (NN / KSPLIT);
  int k1 = k0 + (NN / KSPLIT);
  v8f a0 = {}, a1 = {}, a2 = {}, a3 = {};
#pragma unroll 4
  for (int k = k0; k < k1; k += 4) {
    int kk = k + khalf * 2;
    v2f a  = *(const v2f*)(xr  + kk);
    v2f b0 = *(const v2f*)(b0p + kk);
    v2f b1 = *(const v2f*)(b1p + kk);
    v2f b2 = *(const v2f*)(b2p + kk);
    v2f b3 = *(const v2f*)(b3p + kk);
    a0 = wmma_f32_16x16x4(a, b0, a0);
    a1 = wmma_f32_16x16x4(a, b1, a1);
    a2 = wmma_f32_16x16x4(a, b2, a2);
    a3 = wmma_f32_16x16x4(a, b3, a3);
  }
  float* dst = xhp + (size_t)kc * ((size_t)NB * CIN * NMODE);
#pragma unroll
  for (int v = 0; v < 8; ++v) {
    int rr = r0 + v + 8 * khalf;   // C/D layout: M = v (+8 for upper halfwave)
    float* drow = dst + (size_t)rr * NMODE + mrow;
    drow[0]  = a0[v];
    drow[16] = a1[v];
    drow[32] = a2[v];
    drow[48] = a3[v];
  }
}

// Sum the KSPLIT partial buffers (deterministic, no float atomics)
__global__ void reduce_xh(const float* __restrict__ xhp, float* __restrict__ xh) {
  int idx = blockIdx.x * blockDim.x + threadIdx.x;
  if (idx >= NB * CIN * NMODE) return;
  float s = 0.0f;
#pragma unroll
  for (int c = 0; c < KSPLIT; ++c)
    s += xhp[(size_t)c * ((size_t)NB * CIN * NMODE) + idx];
  xh[idx] = s;
}

// ---------------------------------------------------------------------------
// Mode mixing: per mode k,
//   OutM[b,o,k] = 0.5 * ( XH[b,:,k] @ Wp[:,:,k] + XH[b,:,-k] @ Wm[:,:,k] )
// One wave = one (k, 16-batch, 16-out) tile, K loop over CIN.
// ---------------------------------------------------------------------------
__global__ void mode_mix(const float* __restrict__ xh, const float* __restrict__ wp,
                         const float* __restrict__ wm, float* __restrict__ outm) {
  int wave = (blockIdx.x * blockDim.x + threadIdx.x) >> 5;
  int lane = threadIdx.x & 31;
  int k   = wave >> 5;             // 64 modes
  int rem = wave & 31;
  int bt = rem >> 3;               // 4 batch tiles
  int ot = rem & 7;                // 8 out-channel tiles
  int kneg = (NMODE - k) & (NMODE - 1);
  int mrow = lane & 15;
  int khalf = lane >> 4;
  int bglob = bt * 16 + mrow;      // A row (batch)
  int oglob = ot * 16 + mrow;      // B col (out channel)
  const float* xr = xh + (size_t)bglob * CIN * NMODE;
  v8f acc = {};
#pragma unroll 4
  for (int i = 0; i < CIN; i += 4) {
    int i0 = i + khalf * 2;
    v2f ap, an, bp, bm;
    ap.x = xr[(size_t)i0 * NMODE + k];
    ap.y = xr[(size_t)(i0 + 1) * NMODE + k];
    an.x = xr[(size_t)i0 * NMODE + kneg];
    an.y = xr[(size_t)(i0 + 1) * NMODE + kneg];
    bp.x = wp[((size_t)i0 * COUT + oglob) * NMODE + k];
    bp.y = wp[((size_t)(i0 + 1) * COUT + oglob) * NMODE + k];
    bm.x = wm[((size_t)i0 * COUT + oglob) * NMODE + k];
    bm.y = wm[((size_t)(i0 + 1) * COUT + oglob) * NMODE + k];
    acc = wmma_f32_16x16x4(ap, bp, acc);
    acc = wmma_f32_16x16x4(an, bm, acc);
  }
#pragma unroll
  for (int v = 0; v < 8; ++v) {
    int bb = bt * 16 + v + 8 * khalf;
    outm[((size_t)bb * COUT + oglob) * NMODE + k] = 0.5f * acc[v];
  }
}

// ---------------------------------------------------------------------------
// Inverse: Y[r][n] = sum_{k<64} OutM[r][k] * c2t[n][k]   (c2t includes 1/NN)
// One wave = 16 rows x 64 output columns; writes 4 KiB of the 128 MiB output.
// ---------------------------------------------------------------------------
__global__ void dht_inv(const float* __restrict__ outm, const float* __restrict__ c2t,
                        float* __restrict__ y) {
  int wave = (blockIdx.x * blockDim.x + threadIdx.x) >> 5;
  int lane = threadIdx.x & 31;
  int rt = wave >> 6;              // 512 row tiles
  int nc = wave & 63;              // 64 column chunks of 64
  int r0 = rt * 16;
  int n0 = nc * 64;
  int mrow = lane & 15;
  int khalf = lane >> 4;
  const float* ar  = outm + (size_t)(r0 + mrow) * NMODE;
  const float* b0p = c2t + (size_t)(n0 + mrow)      * NMODE;
  const float* b1p = c2t + (size_t)(n0 + 16 + mrow) * NMODE;
  const float* b2p = c2t + (size_t)(n0 + 32 + mrow) * NMODE;
  const float* b3p = c2t + (size_t)(n0 + 48 + mrow) * NMODE;
  v8f a0 = {}, a1 = {}, a2 = {}, a3 = {};
#pragma unroll 4
  for (int k = 0; k < NMODE; k += 4) {
    int kk = k + khalf * 2;
    v2f a  = *(const v2f*)(ar  + kk);
    v2f b0 = *(const v2f*)(b0p + kk);
    v2f b1 = *(const v2f*)(b1p + kk);
    v2f b2 = *(const v2f*)(b2p + kk);
    v2f b3 = *(const v2f*)(b3p + kk);
    a0 = wmma_f32_16x16x4(a, b0, a0);
    a1 = wmma_f32_16x16x4(a, b1, a1);
    a2 = wmma_f32_16x16x4(a, b2, a2);
    a3 = wmma_f32_16x16x4(a, b3, a3);
  }
#pragma unroll
  for (int v = 0; v < 8; ++v) {
    int rr = r0 + v + 8 * khalf;
    float* drow = y + (size_t)rr * NN + n0 + mrow;
    drow[0]  = a0[v];
    drow[16] = a1[v];
    drow[32] = a2[v];
    drow[48] = a3[v];
  }
}

// ---------------------------------------------------------------------------
extern "C" void kernel_launch(void* const* d_in, const int* in_sizes, int n_in,
                              void* d_out, int out_size, void* d_ws, size_t ws_size,
                              hipStream_t stream) {
  (void)in_sizes; (void)n_in; (void)out_size; (void)ws_size;
  const float* x = (const float*)d_in[0];     // (64, 128, 4096)
  const float* w = (const float*)d_in[1];     // (128, 128, 64)
  float* y = (float*)d_out;                   // (64, 128, 4096)

  // Workspace layout (floats); total = 7,864,320 floats = 30 MiB
  float* ws   = (float*)d_ws;
  float* c1t  = ws;                                         //   64*4096
  float* c2t  = c1t + (size_t)NMODE * NN;                   // 4096*64
  float* wp   = c2t + (size_t)NN * NMODE;                   // 128*128*64
  float* wm   = wp  + (size_t)CIN * COUT * NMODE;           // 128*128*64
  float* xhp  = wm  + (size_t)CIN * COUT * NMODE;           // 8*8192*64
  float* xh   = xhp + (size_t)KSPLIT * NB * CIN * NMODE;    // 8192*64
  float* outm = xh  + (size_t)NB * CIN * NMODE;             // 8192*64

  prep_cas<<<(2 * NMODE * NN) / 256, 256, 0, stream>>>(c1t, c2t);
  prep_w<<<(CIN * COUT * NMODE) / 256, 256, 0, stream>>>(w, wp, wm);

  // 512 row tiles * 8 K chunks = 4096 waves
  dht_fwd<<<((NB * CIN / 16) * KSPLIT * 32) / 256, 256, 0, stream>>>(x, c1t, xhp);
  reduce_xh<<<(NB * CIN * NMODE) / 256, 256, 0, stream>>>(xhp, xh);

  // 64 modes * 4 batch tiles * 8 out tiles = 2048 waves
  mode_mix<<<(NMODE * (NB / 16) * (COUT / 16) * 32) / 256, 256, 0, stream>>>(
      xh, wp, wm, outm);

  // 512 row tiles * 64 column chunks = 32768 waves
  dht_inv<<<((NB * COUT / 16) * (NN / 64) * 32) / 256, 256, 0, stream>>>(
      outm, c2t, y);
}